// FBPReconstructor_14027363189342
// MI455X (gfx1250) — compile-verified
//
#include <hip/hip_runtime.h>
#include <hip/hip_bf16.h>

// ---------------------------------------------------------------------------
// FBP reconstructor for MI455X (gfx1250, wave32).
//
// Pipeline (all linear-algebra identities, exact in fp32):
//   proj_sim[b,a,w] = sum_k Amat[a,k] * F_b[k,w]       (conv+rowsum collapsed)
//   filtered        = (Amat @ (F_b circularly convolved with c))  via WMMA f32
//   out             = backprojection gather-reduce from LDS-staged rows
// ---------------------------------------------------------------------------

typedef float v2f __attribute__((ext_vector_type(2)));
typedef float v8f __attribute__((ext_vector_type(8)));

#define IMG   256
#define DET   256
#define NANG  180
#define MPAD  192   // 180 angles padded to 12 WMMA M-tiles
#define KPAD  12    // 9 features padded to 3 k-steps of V_WMMA_F32_16X16X4_F32

static constexpr float PI_F = 3.14159265358979323846f;

// --- Kernel 1: c[j] = irfft(ramlak)[j]  (256-tap circular filter kernel) ----
__global__ void fbp_build_c(const float* __restrict__ ramlak, float* __restrict__ c) {
    int j = threadIdx.x;                     // 256 threads
    float acc = ramlak[0];                   // k = 0 term
    const float w0 = (2.0f * PI_F) / 256.0f;
    for (int k = 1; k < 128; ++k)
        acc += 2.0f * ramlak[k] * cosf(w0 * (float)(k * j));
    acc += ramlak[128] * cosf(PI_F * (float)j);   // Nyquist term
    c[j] = acc * (1.0f / 256.0f);
}

// --- Kernel 2: zero-padded weight matrix Amat[192][12] ----------------------
__global__ void fbp_build_amat(const float* __restrict__ conv_w, float* __restrict__ Amat) {
    for (int idx = threadIdx.x; idx < MPAD * KPAD; idx += blockDim.x) {
        int a = idx / KPAD, k = idx % KPAD;
        Amat[idx] = (a < NANG && k < 9) ? conv_w[a * 9 + k] : 0.0f;
    }
}

// --- Kernel 3: 9 boundary-corrected column-sum features per batch -----------
// F[b][i*3+j][w] = S_i[w+j-1] (zero outside), S0=colsum-row255, S1=colsum,
// S2=colsum-row0.  Rows 9..11 zero-padded for the WMMA K dimension.
__global__ void fbp_build_features(const float* __restrict__ x, float* __restrict__ FRaw) {
    int b = blockIdx.x;
    int w = threadIdx.x;                     // 256 threads
    __shared__ float S0[DET], S1[DET], S2[DET];
    const float* xb = x + (size_t)b * IMG * IMG;
    float cs = 0.0f;
    for (int h = 0; h < IMG; ++h) cs += xb[h * IMG + w];
    float r0 = xb[w], r255 = xb[255 * IMG + w];
    S0[w] = cs - r255;  S1[w] = cs;  S2[w] = cs - r0;
    __syncthreads();
    float* F = FRaw + (size_t)b * KPAD * DET;
#pragma unroll
    for (int j = 0; j < 3; ++j) {
        int  wj    = w + j - 1;
        bool valid = (wj >= 0) && (wj < DET);
        F[(0 * 3 + j) * DET + w] = valid ? S0[wj] : 0.0f;
        F[(1 * 3 + j) * DET + w] = valid ? S1[wj] : 0.0f;
        F[(2 * 3 + j) * DET + w] = valid ? S2[wj] : 0.0f;
    }
    F[ 9 * DET + w] = 0.0f;
    F[10 * DET + w] = 0.0f;
    F[11 * DET + w] = 0.0f;
}

// --- Kernel 4: Ram-Lak filter each feature row (circular conv with c) -------
__global__ void fbp_filter_features(const float* __restrict__ FRaw,
                                    const float* __restrict__ c,
                                    float* __restrict__ FF) {
    int bk = blockIdx.x;                     // 4*12 rows
    int n  = threadIdx.x;                    // 256 threads
    __shared__ float p[DET], ck[DET];
    p[n]  = FRaw[(size_t)bk * DET + n];
    ck[n] = c[n];
    __syncthreads();
    float acc = 0.0f;
    for (int m = 0; m < DET; ++m)
        acc += p[m] * ck[(n - m) & 255];
    FF[(size_t)bk * DET + n] = acc;
}

// --- Kernel 5: WMMA matmul  filtered[b] = Amat(192x12) @ FF_b(12x256) -------
// One wave (32 threads, EXEC all-ones) per 16x32 output patch: two 16x16
// N-tiles, 3 k-steps of V_WMMA_F32_16X16X4_F32 each -> 6 static v_wmma.
__global__ void __launch_bounds__(32)
fbp_wmma_project(const float* __restrict__ Amat,
                 const float* __restrict__ FF,
                 float* __restrict__ filtered) {
    int wave = blockIdx.x;                   // 4 * 12 * 8 = 384 waves
    int b    = wave / 96;
    int rem  = wave % 96;
    int mt   = rem / 8;                      // 16-row M tile
    int np   = rem % 8;                      // 32-col N patch
    int lane = threadIdx.x;
    int half = lane >> 4;                    // 0: K={0,1}, 1: K={2,3}
    int l    = lane & 15;

    const float* FFb = FF + (size_t)b * KPAD * DET;
    int arow = mt * 16 + l;                  // A-matrix row (both lane halves)
    int col0 = np * 32 + l;                  // first N tile column
    int col1 = col0 + 16;                    // second N tile column

    v8f acc0 = {}; v8f acc1 = {};
#pragma unroll
    for (int s = 0; s < 3; ++s) {
        int k0 = 4 * s + (half ? 2 : 0);
        v2f a;
        a.x = Amat[arow * KPAD + k0];
        a.y = Amat[arow * KPAD + k0 + 1];
        v2f b0, b1;
        b0.x = FFb[(k0    ) * DET + col0];
        b0.y = FFb[(k0 + 1) * DET + col0];
        b1.x = FFb[(k0    ) * DET + col1];
        b1.y = FFb[(k0 + 1) * DET + col1];
        acc0 = __builtin_amdgcn_wmma_f32_16x16x4_f32(false, a, false, b0,
                                                     (short)0, acc0, false, false);
        acc1 = __builtin_amdgcn_wmma_f32_16x16x4_f32(false, a, false, b1,
                                                     (short)0, acc1, false, false);
    }

    float* out = filtered + (size_t)b * MPAD * DET;
#pragma unroll
    for (int v = 0; v < 8; ++v) {
        int row = mt * 16 + (half ? v + 8 : v);
        out[row * DET + col0] = acc0[v];
        out[row * DET + col1] = acc1[v];
    }
}

// --- Kernel 6: back-projection ---------------------------------------------
// One block per (batch, image row). Detector row for each angle staged in LDS.
__global__ void fbp_backproject(const float* __restrict__ filtered,
                                float* __restrict__ out) {
    int blk = blockIdx.x;                    // 4 * 256 blocks
    int b   = blk >> 8;
    int i   = blk & 255;                     // image row  -> yy = lin[i]
    int j   = threadIdx.x;                   // image col  -> xx = lin[j]

    __shared__ float row[DET];
    __shared__ float cs[NANG], sn[NANG];
    if (j < NANG) {
        float ang = (float)j * (PI_F / 179.0f);   // linspace(0, pi, 180) inclusive
        cs[j] = cosf(ang);
        sn[j] = sinf(ang);
    }
    float yv = -1.0f + (2.0f / 255.0f) * (float)i;
    float xv = -1.0f + (2.0f / 255.0f) * (float)j;
    const float* fb = filtered + (size_t)b * MPAD * DET;

    float acc = 0.0f;
    __syncthreads();
    for (int a = 0; a < NANG; ++a) {
        row[j] = fb[a * DET + j];
        __syncthreads();
        float t  = cs[a] * xv + sn[a] * yv;
        float ix = (t + 1.0f) * 0.5f * 255.0f;
        ix = fminf(fmaxf(ix, 0.0f), 255.0f);
        int   i0 = (int)ix;                  // floor (ix >= 0)
        float fr = ix - (float)i0;
        int   i1 = (i0 < 255) ? i0 + 1 : 255;
        float v0 = row[i0];
        float v1 = row[i1];
        acc += v0 + fr * (v1 - v0);
        __syncthreads();
    }
    out[(size_t)b * (IMG * IMG) + i * IMG + j] = acc * (PI_F / (float)NANG);
}

// ---------------------------------------------------------------------------
extern "C" void kernel_launch(void* const* d_in, const int* in_sizes, int n_in,
                              void* d_out, int out_size, void* d_ws, size_t ws_size,
                              hipStream_t stream) {
    (void)in_sizes; (void)n_in; (void)out_size; (void)ws_size;
    const float* x      = (const float*)d_in[0];   // (4,1,256,256)
    const float* conv_w = (const float*)d_in[1];   // (180,1,3,3)
    const float* ramlak = (const float*)d_in[2];   // (129,)
    float* out = (float*)d_out;                    // (4,1,256,256)

    float* ws       = (float*)d_ws;
    float* c        = ws;                               // 256
    float* Amat     = c + DET;                          // 192*12   = 2304
    float* FRaw     = Amat + MPAD * KPAD;               // 4*12*256 = 12288
    float* FF       = FRaw + 4 * KPAD * DET;            // 12288
    float* filtered = FF + 4 * KPAD * DET;              // 4*192*256 = 196608

    fbp_build_c        <<<1,    256, 0, stream>>>(ramlak, c);
    fbp_build_amat     <<<1,    256, 0, stream>>>(conv_w, Amat);
    fbp_build_features <<<4,    256, 0, stream>>>(x, FRaw);
    fbp_filter_features<<<48,   256, 0, stream>>>(FRaw, c, FF);
    fbp_wmma_project   <<<384,  32,  0, stream>>>(Amat, FF, filtered);
    fbp_backproject    <<<1024, 256, 0, stream>>>(filtered, out);
}